// MarginLoss_45526653337924
// MI455X (gfx1250) — compile-verified
//
#include <hip/hip_runtime.h>

// MarginLoss: out[i] = min(max_{j != label[i]} logits[i,j] - logits[i,label[i]], 0)
// B = 65536 rows, C = 1000 fp32 cols. Pure HBM-stream (262 MB): ~11.3 us floor
// at 23.3 TB/s. One wave32 per row; CDNA5 async global->LDS B128 DMA path.
// Label exclusion done by poisoning buf[lab] with -inf in LDS (lane 0 ds_store),
// so the max loop is pure v_max with no per-element compares.

typedef float v4f __attribute__((ext_vector_type(4)));
typedef int   v4i __attribute__((vector_size(16)));   // matches builtin param type

#define MK_C    1000
#define MK_C4   250          // C / 4 (exact)
#define WAVES   8
#define BLOCK   (WAVES * 32)

#if defined(__gfx1250__) && __has_builtin(__builtin_amdgcn_global_load_async_to_lds_b128)
#define USE_ASYNC_LDS 1
#else
#define USE_ASYNC_LDS 0
#endif

#if USE_ASYNC_LDS
typedef __attribute__((address_space(1))) v4i* as1_v4i;
typedef __attribute__((address_space(3))) v4i* as3_v4i;
#endif

__global__ __launch_bounds__(BLOCK) void margin_row_max_kernel(
    const float* __restrict__ logits, const int* __restrict__ label,
    float* __restrict__ out, int nrows) {
  const int lane = threadIdx.x & 31;
  const int wave = threadIdx.x >> 5;
  const int row  = blockIdx.x * WAVES + wave;
  if (row >= nrows) return;

  const float* rowG = logits + (size_t)row * MK_C;
  const int lab = label[row];
  const float NEG_INF = -__builtin_inff();
  float m = NEG_INF;
  float labVal;

#if USE_ASYNC_LDS
  __shared__ alignas(16) float lds[WAVES * MK_C];
  float* buf = &lds[wave * MK_C];
  {
    // Per-lane global/LDS base (16 B per lane); imm offset walks 512 B
    // (32 lanes x 16 B) per issue. 7 full issues + 1 partial (lanes 0..25
    // cover j4 = 224..249).
    as1_v4i gp = (as1_v4i)rowG + lane;
    as3_v4i lp = (as3_v4i)buf + lane;
    __builtin_amdgcn_global_load_async_to_lds_b128(gp, lp, 0,    0);
    __builtin_amdgcn_global_load_async_to_lds_b128(gp, lp, 512,  0);
    __builtin_amdgcn_global_load_async_to_lds_b128(gp, lp, 1024, 0);
    __builtin_amdgcn_global_load_async_to_lds_b128(gp, lp, 1536, 0);
    __builtin_amdgcn_global_load_async_to_lds_b128(gp, lp, 2048, 0);
    __builtin_amdgcn_global_load_async_to_lds_b128(gp, lp, 2560, 0);
    __builtin_amdgcn_global_load_async_to_lds_b128(gp, lp, 3072, 0);
    if (lane < MK_C4 - 7 * 32)
      __builtin_amdgcn_global_load_async_to_lds_b128(gp, lp, 3584, 0);
  }
#if __has_builtin(__builtin_amdgcn_s_wait_asynccnt)
  __builtin_amdgcn_s_wait_asynccnt(0);
#else
  asm volatile("s_wait_asynccnt 0x0" ::: "memory");
#endif
  asm volatile("" ::: "memory");   // keep LDS accesses after the wait

  // Broadcast-read the label logit, then poison its LDS slot with -inf so the
  // max loop needs no per-element label compares. Same-wave LDS ops are
  // in-order (DScnt), and the store may alias the vector loads, so the
  // compiler keeps program order.
  labVal = buf[lab];               // ds_load_b32 broadcast
  if (lane == 0) buf[lab] = NEG_INF;

  #pragma unroll
  for (int k = 0; k < 8; ++k) {
    int j4 = lane + 32 * k;
    if (j4 < MK_C4) {
      v4f v = *(const v4f*)(buf + j4 * 4);   // ds_load_b128
      m = fmaxf(m, fmaxf(fmaxf(v.x, v.y), fmaxf(v.z, v.w)));
    }
  }
#else
  // Fallback: direct streaming B128 loads with non-temporal hint.
  labVal = rowG[lab];
  #pragma unroll
  for (int k = 0; k < 8; ++k) {
    int j4 = lane + 32 * k;
    if (j4 < MK_C4) {
      v4f v = __builtin_nontemporal_load((const v4f*)rowG + j4);
      int c0 = j4 * 4;
      float a = (c0 + 0 == lab) ? NEG_INF : v.x;
      float b = (c0 + 1 == lab) ? NEG_INF : v.y;
      float c = (c0 + 2 == lab) ? NEG_INF : v.z;
      float d = (c0 + 3 == lab) ? NEG_INF : v.w;
      m = fmaxf(m, fmaxf(fmaxf(a, b), fmaxf(c, d)));
    }
  }
#endif

  // Wave32 max reduction (all 32 lanes active here).
  #pragma unroll
  for (int off = 16; off > 0; off >>= 1)
    m = fmaxf(m, __shfl_xor(m, off, 32));

  if (lane == 0)
    out[row] = fminf(m - labVal, 0.0f);   // == -relu(labVal - max_other)
}

extern "C" void kernel_launch(void* const* d_in, const int* in_sizes, int n_in,
                              void* d_out, int out_size, void* d_ws, size_t ws_size,
                              hipStream_t stream) {
  const float* logits = (const float*)d_in[0];
  const int*   label  = (const int*)d_in[1];
  float*       out    = (float*)d_out;
  (void)d_ws; (void)ws_size; (void)n_in; (void)out_size;

  int nrows  = in_sizes[1];                 // B (label has one entry per row)
  int blocks = (nrows + WAVES - 1) / WAVES; // one wave per row
  margin_row_max_kernel<<<blocks, BLOCK, 0, stream>>>(logits, label, out, nrows);
}